// LightGCN_35845797053213
// MI455X (gfx1250) — compile-verified
//
#include <hip/hip_runtime.h>
#include <hip/hip_bf16.h>
#include <math.h>

typedef float v2f __attribute__((ext_vector_type(2)));
typedef float v8f __attribute__((ext_vector_type(8)));

#define EMB 16
#define NUM_LAYERS 3

// Hardware f32 atomic add (global_atomic_add_f32, non-returning path).
__device__ __forceinline__ void atom_add_f32(float* p, float v) {
    unsafeAtomicAdd(p, v);   // HIP: guaranteed native FP atomic on AMD
}

// ---------------------------------------------------------------------------
// init: x_a = emb_table[node_label_index], final = x_a, cnt = 0, x_b = 0
// ---------------------------------------------------------------------------
__global__ void lgcn_init(const float* __restrict__ emb,
                          const int* __restrict__ nli,
                          float* __restrict__ xa,
                          float* __restrict__ xb,
                          float* __restrict__ fin,
                          float* __restrict__ cnt,
                          int N) {
    int id = blockIdx.x * blockDim.x + threadIdx.x;
    if (id < N) cnt[id] = 0.0f;
    int total = N * EMB;
    if (id < total) {
        int i = id >> 4;
        int j = id & 15;
        float v = emb[(long)nli[i] * EMB + j];
        xa[id]  = v;
        xb[id]  = 0.0f;        // layer-0 accumulator
        fin[id] = v;
    }
}

// ---------------------------------------------------------------------------
// count: in-degree via f32 atomics (layer-invariant; computed once)
// ---------------------------------------------------------------------------
__global__ void lgcn_count(const int* __restrict__ dst,
                           float* __restrict__ cnt, int E) {
    int e = blockIdx.x * blockDim.x + threadIdx.x;
    if (e >= E) return;
    int d = __builtin_nontemporal_load(dst + e);   // streamed once: NT hint
    atom_add_f32(cnt + d, 1.0f);
}

__global__ void lgcn_inv(float* __restrict__ cnt, int N) {
    int i = blockIdx.x * blockDim.x + threadIdx.x;
    if (i >= N) return;
    float c = cnt[i];
    cnt[i] = 1.0f / (c > 1.0f ? c : 1.0f);
}

// ---------------------------------------------------------------------------
// scatter: one lane per edge; 64B row gather (L2-resident), 16 f32 L2 atomics
// ---------------------------------------------------------------------------
__global__ void lgcn_scatter(const float* __restrict__ xcur,
                             float* __restrict__ xnxt,
                             const int* __restrict__ src,
                             const int* __restrict__ dst,
                             int E) {
    int e = blockIdx.x * blockDim.x + threadIdx.x;
    if (e >= E) return;
    int s = __builtin_nontemporal_load(src + e);   // index streams: HBM, NT
    int d = __builtin_nontemporal_load(dst + e);
    const float4* r = (const float4*)(xcur + (long)s * EMB);
    float4 r0 = r[0], r1 = r[1], r2 = r[2], r3 = r[3];
    float* w = xnxt + (long)d * EMB;
    atom_add_f32(w + 0,  r0.x); atom_add_f32(w + 1,  r0.y);
    atom_add_f32(w + 2,  r0.z); atom_add_f32(w + 3,  r0.w);
    atom_add_f32(w + 4,  r1.x); atom_add_f32(w + 5,  r1.y);
    atom_add_f32(w + 6,  r1.z); atom_add_f32(w + 7,  r1.w);
    atom_add_f32(w + 8,  r2.x); atom_add_f32(w + 9,  r2.y);
    atom_add_f32(w + 10, r2.z); atom_add_f32(w + 11, r2.w);
    atom_add_f32(w + 12, r3.x); atom_add_f32(w + 13, r3.y);
    atom_add_f32(w + 14, r3.z); atom_add_f32(w + 15, r3.w);
}

// ---------------------------------------------------------------------------
// normalize: x_next *= 1/deg (in place -> next layer's input), final += x_next,
//            and zero the just-consumed ping buffer (next layer's accumulator)
// ---------------------------------------------------------------------------
__global__ void lgcn_norm(float* __restrict__ nxt,
                          float* __restrict__ fin,
                          const float* __restrict__ inv,
                          float* __restrict__ tozero,
                          int N) {
    int id = blockIdx.x * blockDim.x + threadIdx.x;
    if (id >= N * EMB) return;
    float s = inv[id >> 4];
    float v = nxt[id] * s;
    nxt[id] = v;
    fin[id] += v;
    tozero[id] = 0.0f;
}

// ---------------------------------------------------------------------------
// score: 16 edges per wave via V_WMMA_F32_16X16X4_F32.
//   A[m][k] = final[n1[base+m]][k]   (lane m / m+16 holds k = 2*half+{0,1})
//   B[k][n] = final[n2[base+n]][k]   (lane n / n+16 holds rows k = 2*half+{0,1})
//   D[i][i] = dot(n1_i, n2_i); 4 chained k=4 WMMAs cover K=16.
//   out = sigmoid(D[i][i] / NUM_LAYERS^2)   (folds the final/3 scaling)
// ---------------------------------------------------------------------------
__global__ void lgcn_score(const float* __restrict__ fin,
                           const int* __restrict__ n1,
                           const int* __restrict__ n2,
                           float* __restrict__ out,
                           int L) {
    __shared__ float lds[8 * 256];                  // 8 waves * 32 lanes * 8 f32
    int gtid = blockIdx.x * blockDim.x + threadIdx.x;
    int wave = gtid >> 5;
    int lane = threadIdx.x & 31;
    int wib  = threadIdx.x >> 5;
    int base = wave * 16;
    if (base >= L) return;

    int col  = lane & 15;       // edge-in-tile (A row index == B col index)
    int half = lane >> 4;       // which K pair this lane holds
    int e = base + col;
    if (e >= L) e = L - 1;      // tail tile: duplicate last edge (harmless)
    long p1 = (long)n1[e] * EMB;
    long p2 = (long)n2[e] * EMB;

    v8f acc = {};
#pragma unroll
    for (int c = 0; c < 4; ++c) {
        v2f a = *(const v2f*)(fin + p1 + 4 * c + 2 * half);
        v2f b = *(const v2f*)(fin + p2 + 4 * c + 2 * half);
        // 8 args: (neg_a, A, neg_b, B, c_mod, C, reuse_a, reuse_b)
        acc = __builtin_amdgcn_wmma_f32_16x16x4_f32(
            false, a, false, b, (short)0, acc, false, false);
    }

    // Spill D through LDS and pull the diagonal.
    float* myl = lds + wib * 256 + lane * 8;
#pragma unroll
    for (int v = 0; v < 8; ++v) myl[v] = acc[v];
    __syncthreads();

    if (lane < 16 && base + lane < L) {
        int m = lane;
        int srcLane = (m < 8) ? m : (m + 16);       // D[m][m] location
        int vv = m & 7;
        float d = lds[wib * 256 + srcLane * 8 + vv];
        d *= (1.0f / (float)(NUM_LAYERS * NUM_LAYERS));
        // fast sigmoid: native exp + v_rcp_f32 (score stage is L2-bound anyway)
        float t = __expf(-d);
        float sig = __builtin_amdgcn_rcpf(1.0f + t);
        __builtin_nontemporal_store(sig, out + base + m);
    }
}

// ---------------------------------------------------------------------------
extern "C" void kernel_launch(void* const* d_in, const int* in_sizes, int n_in,
                              void* d_out, int out_size, void* d_ws, size_t ws_size,
                              hipStream_t stream) {
    const float* emb   = (const float*)d_in[0];
    const int*   eidx  = (const int*)d_in[1];   // [2, E] row-major
    const int*   elidx = (const int*)d_in[2];   // [2, L] row-major
    const int*   nli   = (const int*)d_in[3];

    int N = in_sizes[3];
    int E = in_sizes[1] / 2;
    int L = in_sizes[2] / 2;

    const int* src  = eidx;
    const int* dstv = eidx + E;
    const int* n1   = elidx;
    const int* n2   = elidx + L;

    // workspace layout: cnt[N] | x_a[N*16] | x_b[N*16] | final[N*16]
    float* ws  = (float*)d_ws;
    float* cnt = ws;
    float* xa  = cnt + N;
    float* xb  = xa + (size_t)N * EMB;
    float* fin = xb + (size_t)N * EMB;
    float* out = (float*)d_out;

    const int BLK = 256;
    int ne = N * EMB;

    lgcn_init <<<(ne + BLK - 1) / BLK, BLK, 0, stream>>>(emb, nli, xa, xb, fin, cnt, N);
    lgcn_count<<<(E  + BLK - 1) / BLK, BLK, 0, stream>>>(dstv, cnt, E);
    lgcn_inv  <<<(N  + BLK - 1) / BLK, BLK, 0, stream>>>(cnt, N);

    float* cur = xa;
    float* nxt = xb;
    for (int k = 0; k < NUM_LAYERS; ++k) {
        lgcn_scatter<<<(E  + BLK - 1) / BLK, BLK, 0, stream>>>(cur, nxt, src, dstv, E);
        // normalize nxt, accumulate into final, and zero cur (next layer's acc)
        lgcn_norm   <<<(ne + BLK - 1) / BLK, BLK, 0, stream>>>(nxt, fin, cnt, cur, N);
        float* t = cur; cur = nxt; nxt = t;
    }

    int waves   = (L + 15) / 16;
    int threads = waves * 32;
    lgcn_score<<<(threads + BLK - 1) / BLK, BLK, 0, stream>>>(fin, n1, n2, out, L);
}